// ViscoplasticMaterialModel_34333968564382
// MI455X (gfx1250) — compile-verified
//
#include <hip/hip_runtime.h>
#include <stdint.h>

#define NIV      8
#define T_STEPS  256
#define DT_C     0.01f
#define TB       64                      // batches per block
#define TC       16                      // timesteps per LDS chunk
#define NCHUNK   (T_STEPS / TC)          // 16
#define ROW_BYTES   (TC * NIV * 4)       // 512 B contiguous per batch per chunk
#define PAD_BYTES   32                   // TDM pad: 8 DWORDs per 128 DWORDs
#define ROW_STRIDE  (ROW_BYTES + PAD_BYTES)   // 544 B -> row offset % 64 banks = 8
#define BUF_BYTES   (TB * ROW_STRIDE)    // 34816 B per buffer

typedef unsigned int u32x4 __attribute__((ext_vector_type(4)));
typedef int          i32x4 __attribute__((ext_vector_type(4)));
typedef int          i32x8 __attribute__((ext_vector_type(8)));

// TDM 2D tile load: TB rows of ROW_BYTES, global row stride = T*NIV elements,
// packed into LDS with 32B padding every 512B (bank-conflict-free compute reads).
__device__ __forceinline__ void tdm_load_tile(uint32_t lds_byte_off, uint64_t gaddr_bytes) {
  u32x4 g0;
  g0[0] = 1u;                                        // count=1, is_restore=0, gather off
  g0[1] = lds_byte_off;                              // lds_addr [63:32]
  g0[2] = (uint32_t)(gaddr_bytes & 0xFFFFFFFFu);     // global_addr[31:0]  -> bits [95:64]
  g0[3] = (uint32_t)((gaddr_bytes >> 32) & 0x01FFFFFFu)  // global_addr[56:32] -> [120:96]
        | (2u << 30);                                // type=2 ("image")
  i32x8 g1;
  g1[0] = (2 << 16)                                  // data_size = 4 B
        | (1 << 20)                                  // pad_enable
        | (6 << 22)                                  // pad_interval: 128 DWORDs (512 B)
        | (7 << 25);                                 // pad_amount:   8 DWORDs (32 B)
  g1[1] = (int)(((uint32_t)(TC * NIV)) << 16);       // tensor_dim0[15:0] at bits [63:48]
  g1[2] = (int)(((uint32_t)TB) << 16);               // tensor_dim1[15:0] at bits [95:80]
  g1[3] = (int)(((uint32_t)(TC * NIV)) << 16);       // tile_dim0 at bits [127:112]
  g1[4] = (int)TB;                                   // tile_dim1 at [143:128]; tile_dim2=0
  g1[5] = (int)(T_STEPS * NIV);                      // tensor_dim0_stride[31:0] = 2048 elems
  g1[6] = 0;
  g1[7] = 0;
  i32x4 gz = {0, 0, 0, 0};
#if __clang_major__ >= 23
  i32x8 gz8 = {0, 0, 0, 0, 0, 0, 0, 0};
  __builtin_amdgcn_tensor_load_to_lds(g0, g1, gz, gz, gz8, 0);
#else
  __builtin_amdgcn_tensor_load_to_lds(g0, g1, gz, gz, 0);
#endif
}

__global__ __launch_bounds__(256) void visco_scan_kernel(
    const float* __restrict__ e, const float* __restrict__ E,
    const float* __restrict__ Y, const float* __restrict__ n,
    const float* __restrict__ edot0, float* __restrict__ out, int B) {
  __shared__ __align__(16) unsigned char smem[2 * BUF_BYTES];

  const int tid = threadIdx.x;
  const int bl  = tid >> 2;                 // batch-in-block 0..63
  const int jp  = tid & 3;                  // pair index: handles j = 2*jp, 2*jp+1
  const int b   = blockIdx.x * TB + bl;

  // m = NIV / sum_j(1/E[b,j]): 2 reciprocals per lane + reduce across 4 lanes.
  const float2 Eb = *(const float2*)(E + (size_t)b * NIV + 2 * jp);
  float s = 1.0f / Eb.x + 1.0f / Eb.y;
  s += __shfl_xor(s, 1, 4);
  s += __shfl_xor(s, 2, 4);
  const float m  = (float)NIV / s;
  const float nn = n[b];
  // yn = edot_0 * Y^{-n} / NIV
  const float yn = edot0[b] * exp2f(-nn * log2f(Y[b])) * (1.0f / (float)NIV);

  const size_t OUT_HALF = (size_t)B * T_STEPS * NIV;
  float* __restrict__ stressO = out;
  float* __restrict__ xiO     = out + OUT_HALF;

  const uint32_t ldsbase = (uint32_t)(uintptr_t)(&smem[0]);
  const uint64_t ebase   = (uint64_t)(uintptr_t)e
                         + (uint64_t)blockIdx.x * TB * T_STEPS * NIV * 4ull;

  // Prime the pipeline: wave 0 drives the TDM (uniform branch; EXEC-independent issue).
  if (tid < 32) tdm_load_tile(ldsbase, ebase);

  float  xi0 = 0.0f, xi1 = 0.0f;
  size_t gidx = (size_t)b * T_STEPS * NIV + 2 * jp;

  for (int c = 0; c < NCHUNK; ++c) {
    if (tid < 32) __builtin_amdgcn_s_wait_tensorcnt(0);   // chunk c resident in LDS
    __syncthreads();                                       // publish; all done with buf[1-cur]
    if (tid < 32 && (c + 1) < NCHUNK)
      tdm_load_tile(ldsbase + (uint32_t)(((c + 1) & 1) * BUF_BYTES),
                    ebase + (uint64_t)(c + 1) * ROW_BYTES);

    const float* __restrict__ lrow =
        (const float*)(smem + (c & 1) * BUF_BYTES + bl * ROW_STRIDE) + 2 * jp;
#pragma unroll
    for (int tl = 0; tl < TC; ++tl) {
      const float2 ev = *(const float2*)(lrow + tl * NIV);     // ds_load_b64, no conflicts
      const float q0 = m * (ev.x - xi0);
      const float q1 = m * (ev.y - xi1);
      const float p0 = exp2f(nn * log2f(fabsf(q0)));           // |q|^n (0 -> 0)
      const float p1 = exp2f(nn * log2f(fabsf(q1)));
      const float k0 = copysignf(p0, q0) * yn;
      const float k1 = copysignf(p1, q1) * yn;
      float2 qv;  qv.x = q0;  qv.y = q1;
      float2 xv;  xv.x = xi0; xv.y = xi1;
      *(float2*)(stressO + gidx) = qv;                         // global_store_b64
      *(float2*)(xiO + gidx)     = xv;                         // pre-update xi (scan carry)
      xi0 += DT_C * k0;
      xi1 += DT_C * k1;
      gidx += NIV;
    }
  }
}

extern "C" void kernel_launch(void* const* d_in, const int* in_sizes, int n_in,
                              void* d_out, int out_size, void* d_ws, size_t ws_size,
                              hipStream_t stream) {
  const float* e     = (const float*)d_in[0];
  const float* E     = (const float*)d_in[1];
  const float* Y     = (const float*)d_in[2];
  const float* n     = (const float*)d_in[3];
  const float* edot0 = (const float*)d_in[4];
  float* out = (float*)d_out;

  const int B = in_sizes[1] / NIV;        // E is [B, NIV]
  dim3 grid(B / TB), block(TB * (NIV / 2));
  hipLaunchKernelGGL(visco_scan_kernel, grid, block, 0, stream,
                     e, E, Y, n, edot0, out, B);
}